// ForMicroLoss_29618094474152
// MI455X (gfx1250) — compile-verified
//
#include <hip/hip_runtime.h>

typedef float f32x4 __attribute__((ext_vector_type(4)));
typedef int   i32x4 __attribute__((ext_vector_type(4)));
typedef float v2f   __attribute__((ext_vector_type(2)));
typedef float v8f   __attribute__((ext_vector_type(8)));

// ---------------------------------------------------------------------------
// Wave32 sum reduction.
// WMMA path: A is the 16x4 f32 matrix striped as
//   VGPR0: lanes0-15 -> A[m][0],  lanes16-31 -> A[m][2]
//   VGPR1: lanes0-15 -> A[m][1],  lanes16-31 -> A[m][3]
// With A = (v, 0) and B = all-ones (layout independent), D[m][n] = v[m]+v[m+16].
// Lane 0 holds D[0..7][0] in its 8 dest VGPRs, lane 16 holds D[8..15][0],
// so summing the 8 dest regs and adding lane0 + lane16 gives the full sum.
// ---------------------------------------------------------------------------
__device__ __forceinline__ float wave_sum32(float v) {
#if defined(__HIP_DEVICE_COMPILE__) && __has_builtin(__builtin_amdgcn_wmma_f32_16x16x4_f32)
    v2f a; a.x = v;    a.y = 0.0f;
    v2f b; b.x = 1.0f; b.y = 1.0f;
    v8f c = {};
    v8f d = __builtin_amdgcn_wmma_f32_16x16x4_f32(
        /*neg_a=*/false, a, /*neg_b=*/false, b,
        /*c_mod=*/(short)0, c, /*reuse_a=*/false, /*reuse_b=*/false);
    float s = d[0] + d[1] + d[2] + d[3] + d[4] + d[5] + d[6] + d[7];
    return __shfl(s, 0, 32) + __shfl(s, 16, 32);
#else
    for (int off = 16; off > 0; off >>= 1) v += __shfl_down(v, off, 32);
    return __shfl(v, 0, 32);
#endif
}

// ---------------------------------------------------------------------------
// Pass 1: grid-stride stream over pred/labels with NT b128 loads.
// Per element: s = log1p(exp(-|p|)); softplus(-p) = s + max(-p,0);
// softplus(p) = s + max(p,0). Accumulate pos-term, neg-term, pos-count.
// Each block writes one float4 partial {pos, neg, cnt, 0} to ws.
// ---------------------------------------------------------------------------
__global__ __launch_bounds__(256) void fml_partial(
    const float* __restrict__ pred, const int* __restrict__ lab,
    f32x4* __restrict__ ws, long long n)
{
    const long long n4     = n >> 2;
    const long long tid    = (long long)blockIdx.x * blockDim.x + threadIdx.x;
    const long long stride = (long long)gridDim.x * blockDim.x;

    float pos = 0.0f, neg = 0.0f, cnt = 0.0f;

    const f32x4* __restrict__ p4 = (const f32x4*)pred;
    const i32x4* __restrict__ y4 = (const i32x4*)lab;

    for (long long j = tid; j < n4; j += stride) {
        f32x4 p = __builtin_nontemporal_load(p4 + j);
        i32x4 y = __builtin_nontemporal_load(y4 + j);
#pragma unroll
        for (int c = 0; c < 4; ++c) {
            float pc = p[c];
            float s  = __logf(1.0f + __expf(-__builtin_fabsf(pc)));
            bool ispos = (y[c] == 1);
            pos += ispos ? (s + __builtin_fmaxf(-pc, 0.0f)) : 0.0f;
            neg += ispos ? 0.0f : (s + __builtin_fmaxf(pc, 0.0f));
            cnt += ispos ? 1.0f : 0.0f;
        }
    }
    // Scalar tail (N*K = 2^26 is divisible by 4, but stay general).
    for (long long j = (n4 << 2) + tid; j < n; j += stride) {
        float pc = pred[j];
        float s  = __logf(1.0f + __expf(-__builtin_fabsf(pc)));
        bool ispos = (lab[j] == 1);
        pos += ispos ? (s + __builtin_fmaxf(-pc, 0.0f)) : 0.0f;
        neg += ispos ? 0.0f : (s + __builtin_fmaxf(pc, 0.0f));
        cnt += ispos ? 1.0f : 0.0f;
    }

    // All lanes reconverged here -> EXEC all-ones, WMMA-legal.
    float wpos = wave_sum32(pos);
    float wneg = wave_sum32(neg);
    float wcnt = wave_sum32(cnt);

    __shared__ float red[3][8];
    const int wave = threadIdx.x >> 5, lane = threadIdx.x & 31;
    if (lane == 0) { red[0][wave] = wpos; red[1][wave] = wneg; red[2][wave] = wcnt; }
    __syncthreads();
    if (threadIdx.x == 0) {
        float P = 0.0f, Ng = 0.0f, C = 0.0f;
#pragma unroll
        for (int i = 0; i < 8; ++i) { P += red[0][i]; Ng += red[1][i]; C += red[2][i]; }
        f32x4 o; o[0] = P; o[1] = Ng; o[2] = C; o[3] = 0.0f;
        __builtin_nontemporal_store(o, ws + blockIdx.x);
    }
}

// ---------------------------------------------------------------------------
// Pass 2: one 256-thread block reduces the per-block partials in f64 and
// emits loss = pos_sum/n_pos + neg_sum/n_neg.
// ---------------------------------------------------------------------------
__global__ __launch_bounds__(256) void fml_final(
    const f32x4* __restrict__ ws, int nblocks,
    float* __restrict__ out, long long n)
{
    double pos = 0.0, neg = 0.0, cnt = 0.0;
    for (int j = threadIdx.x; j < nblocks; j += blockDim.x) {
        f32x4 v = ws[j];
        pos += (double)v[0]; neg += (double)v[1]; cnt += (double)v[2];
    }
    for (int off = 16; off > 0; off >>= 1) {
        pos += __shfl_down(pos, off, 32);
        neg += __shfl_down(neg, off, 32);
        cnt += __shfl_down(cnt, off, 32);
    }
    __shared__ double red[3][8];
    const int wave = threadIdx.x >> 5, lane = threadIdx.x & 31;
    if (lane == 0) { red[0][wave] = pos; red[1][wave] = neg; red[2][wave] = cnt; }
    __syncthreads();
    if (threadIdx.x == 0) {
        double P = 0.0, Ng = 0.0, C = 0.0;
#pragma unroll
        for (int i = 0; i < 8; ++i) { P += red[0][i]; Ng += red[1][i]; C += red[2][i]; }
        double npos = C;
        double nneg = (double)n - C;
        out[0] = (float)(P / npos + Ng / nneg);
    }
}

// ---------------------------------------------------------------------------
// Launch: 2048 blocks x 256 threads (8 waves/block, wave32) = 16K waves,
// each thread streams ~32 b128 loads per array -> enough MLP to saturate HBM.
// Two-pass reduction avoids atomics (workspace is poisoned, never pre-zeroed).
// ---------------------------------------------------------------------------
extern "C" void kernel_launch(void* const* d_in, const int* in_sizes, int n_in,
                              void* d_out, int out_size, void* d_ws, size_t ws_size,
                              hipStream_t stream) {
    (void)n_in; (void)out_size;
    const float* pred = (const float*)d_in[0];
    const int*   lab  = (const int*)d_in[1];
    const long long n = (long long)in_sizes[0];

    const int block = 256;
    int grid = 2048;
    const size_t max_blocks = ws_size / sizeof(f32x4);
    if ((size_t)grid > max_blocks) grid = (int)max_blocks;
    if (grid < 1) grid = 1;

    fml_partial<<<grid, block, 0, stream>>>(pred, lab, (f32x4*)d_ws, n);
    fml_final<<<1, block, 0, stream>>>((const f32x4*)d_ws, grid, (float*)d_out, n);
}